// TitansMemory_21964462752103
// MI455X (gfx1250) — compile-verified
//
#include <hip/hip_runtime.h>
#include <math.h>

#define D_MODEL 512
#define BATCH   8
#define SEQ_T   4096
#define LR      0.01f
#define CLIP    0.5f
#define CHUNK   8

typedef float v2f __attribute__((ext_vector_type(2)));
typedef float v8f __attribute__((ext_vector_type(8)));

// D = A(16x4, f32) * B(4x16, f32) + C(16x16, f32), wave32
static __device__ __forceinline__ v8f wmma4(v2f a, v2f b, v8f c) {
  return __builtin_amdgcn_wmma_f32_16x16x4_f32(false, a, false, b, (short)0, c,
                                               false, false);
}

// ---------------------------------------------------------------------------
// Kernel 1: batch means A[t][d] = mean_b x[b][t][d], and anorm2[t] = ||a_t||^2
// ---------------------------------------------------------------------------
__global__ void __launch_bounds__(512) k_avg(const float* __restrict__ x,
                                             float* __restrict__ A,
                                             float* __restrict__ anorm2) {
  const int t = blockIdx.x;
  const int d = threadIdx.x;
  float s = 0.f;
#pragma unroll
  for (int b = 0; b < BATCH; ++b)
    s += x[((size_t)b * SEQ_T + t) * D_MODEL + d];
  s *= (1.0f / BATCH);
  A[(size_t)t * D_MODEL + d] = s;

  float sq = s * s;
#pragma unroll
  for (int off = 16; off > 0; off >>= 1) sq += __shfl_xor(sq, off);
  __shared__ float red[16];
  if ((d & 31) == 0) red[d >> 5] = sq;
  __syncthreads();
  if (d == 0) {
    float tot = 0.f;
#pragma unroll
    for (int w = 0; w < 16; ++w) tot += red[w];
    anorm2[t] = tot;
  }
}

// ---------------------------------------------------------------------------
// Kernel 2: P0 = A @ W_base^T   (M=4096, N=512, K=512), WMMA f32 16x16x4
// Grid: (M/16, N/64), block = 128 (4 waves, one 16x16 tile per wave)
// ---------------------------------------------------------------------------
__global__ void __launch_bounds__(128) k_p0(const float* __restrict__ A,
                                            const float* __restrict__ Wb,
                                            float* __restrict__ P0) {
  const int lane = threadIdx.x & 31;
  const int wave = threadIdx.x >> 5;
  const int m0 = blockIdx.x * 16;
  const int n0 = blockIdx.y * 64 + wave * 16;
  const int r = lane & 15;
  const int h = lane >> 4;

  const float* arow = A + (size_t)(m0 + r) * D_MODEL;
  const float* brow = Wb + (size_t)(n0 + r) * D_MODEL;  // B[k][n]=Wb[n][k]
  v8f acc = {};
  for (int kb = 0; kb < D_MODEL; kb += 4) {
    v2f a = *(const v2f*)(arow + kb + 2 * h);
    v2f b = *(const v2f*)(brow + kb + 2 * h);
    acc = wmma4(a, b, acc);
  }
#pragma unroll
  for (int v = 0; v < 8; ++v)
    P0[(size_t)(m0 + v + 8 * h) * D_MODEL + n0 + r] = acc[v];
}

// ---------------------------------------------------------------------------
// Kernel 3: chunked sequential scan (C=8). One workgroup, 1024 threads.
// Thread (j = tid>>1, half = tid&1) owns W_c[j][half*256 .. +256) in global ws.
// Per chunk: W_c is read once (8 fused matvecs) and updated once (8 fused
// rank-1 updates); intra-chunk cross terms handled exactly via the 8x8 Gram
// matrix. Emits U[t][:] = LR*scale_t*s_t; writes final W to d_out tail.
// ---------------------------------------------------------------------------
__global__ void __launch_bounds__(1024) k_scan(const float* __restrict__ Wcur,
                                               const float* __restrict__ A,
                                               const float* __restrict__ P0,
                                               const float* __restrict__ anorm2,
                                               float* __restrict__ Wc,
                                               float* __restrict__ U,
                                               float* __restrict__ Wfinal) {
  const int tid = threadIdx.x;
  const int j = tid >> 1;
  const int half = tid & 1;
  const int ibase = half * 256;

  float* wrow = Wc + (size_t)j * D_MODEL + ibase;
  {
    const float* wsrc = Wcur + (size_t)j * D_MODEL + ibase;
    for (int i = 0; i < 256; ++i) wrow[i] = wsrc[i];
  }

  __shared__ float sA[CHUNK * D_MODEL];   // 16 KB: chunk mean vectors
  __shared__ float sM[CHUNK * D_MODEL];   // 16 KB: running m-base per step
  __shared__ float sU[CHUNK * D_MODEL];   // 16 KB: chunk u vectors
  __shared__ float sred[1024];
  __shared__ float sG[CHUNK * CHUNK];     // Gram matrix a_c . a_c'
  __shared__ float sC;

  for (int t0 = 0; t0 < SEQ_T; t0 += CHUNK) {
    __syncthreads();  // protect sA/sU against previous chunk's tail reads
    for (int idx = tid; idx < CHUNK * D_MODEL; idx += 1024)
      sA[idx] = A[(size_t)t0 * D_MODEL + idx];
    __syncthreads();

    // ---- Gram matrix: 64 pairs x 16 threads ----
    {
      const int p = tid >> 4;           // pair id 0..63
      const int l16 = tid & 15;
      const int c = p >> 3, cc = p & 7;
      float g = 0.f;
      for (int i = l16; i < D_MODEL; i += 16)
        g += sA[c * D_MODEL + i] * sA[cc * D_MODEL + i];
      sred[tid] = g;
    }
    __syncthreads();
    if ((tid & 15) == 0) {
      float g = 0.f;
#pragma unroll
      for (int k = 0; k < 16; ++k) g += sred[tid + k];
      sG[tid >> 4] = g;
    }

    // ---- fused matvecs: mb[c] = sum_i Wc[j][i]*a_c[i] (one pass over Wc) ----
    float mb[CHUNK];
#pragma unroll
    for (int c = 0; c < CHUNK; ++c) mb[c] = 0.f;
    for (int i = 0; i < 256; ++i) {
      const float w = wrow[i];
#pragma unroll
      for (int c = 0; c < CHUNK; ++c) mb[c] += w * sA[c * D_MODEL + ibase + i];
    }
    if (half == 0) {
#pragma unroll
      for (int c = 0; c < CHUNK; ++c) sM[c * D_MODEL + j] = mb[c];
    }
    __syncthreads();
    if (half == 1) {
#pragma unroll
      for (int c = 0; c < CHUNK; ++c)
        sM[c * D_MODEL + j] += mb[c] + P0[(size_t)(t0 + c) * D_MODEL + j];
    }
    __syncthreads();

    // ---- inner sequential recurrence over the chunk ----
    for (int c = 0; c < CHUNK; ++c) {
      float sj = 0.f, sq = 0.f;
      if (tid < 512) {
        const float m = sM[c * D_MODEL + tid];
        sj = sA[c * D_MODEL + tid] - m;
        sq = sj * sj;
      }
#pragma unroll
      for (int off = 16; off > 0; off >>= 1) sq += __shfl_xor(sq, off);
      if ((tid & 31) == 0) sred[tid >> 5] = sq;
      __syncthreads();
      if (tid == 0) {
        float tot = 0.f;
#pragma unroll
        for (int w = 0; w < 16; ++w) tot += sred[w];
        const float n = sqrtf(tot * anorm2[t0 + c]);  // ||s|| * ||a||
        sC = LR * ((n > CLIP) ? (CLIP / n) : 1.0f);
      }
      __syncthreads();
      if (tid < 512) {
        const float u = sC * sj;
        sU[c * D_MODEL + tid] = u;
        U[(size_t)(t0 + c) * D_MODEL + tid] = u;
        // propagate this rank-1 update to later steps in the chunk
        for (int c2 = c + 1; c2 < CHUNK; ++c2)
          sM[c2 * D_MODEL + tid] += sG[c2 * CHUNK + c] * u;
      }
      // no barrier needed: next iteration's sM/sA reads are same-thread,
      // and sred writes are fenced by the barrier after sC above.
    }
    __syncthreads();  // sU complete before the Wc update pass

    // ---- fused rank-1 updates: Wc[j][i] += sum_c u_c[j]*a_c[i] ----
    float uj[CHUNK];
#pragma unroll
    for (int c = 0; c < CHUNK; ++c) uj[c] = sU[c * D_MODEL + j];
    for (int i = 0; i < 256; ++i) {
      float acc = 0.f;
#pragma unroll
      for (int c = 0; c < CHUNK; ++c) acc += uj[c] * sA[c * D_MODEL + ibase + i];
      wrow[i] += acc;
    }
  }

  // W_final: each thread writes back exactly the segment it owns
  for (int i = 0; i < 256; ++i)
    Wfinal[(size_t)j * D_MODEL + ibase + i] = wrow[i];
}

// ---------------------------------------------------------------------------
// Kernel 4: causal low-rank attention + base projection + fused gate.
// Grid (T/16, B), block 256 (8 waves). Each block -> out[b, t0:t0+16, :].
// ---------------------------------------------------------------------------
#define XS_STRIDE 514
#define SL_STRIDE 66

__global__ void __launch_bounds__(256) k_attn(const float* __restrict__ x,
                                              const float* __restrict__ Wb,
                                              const float* __restrict__ Wcur,
                                              const float* __restrict__ gw,
                                              const float* __restrict__ gb,
                                              const float* __restrict__ A,
                                              const float* __restrict__ U,
                                              float* __restrict__ out) {
  const int t0 = blockIdx.x * 16;
  const int b = blockIdx.y;
  const int tid = threadIdx.x;
  const int lane = tid & 31;
  const int wave = tid >> 5;  // 0..7
  const int r = lane & 15;
  const int h = lane >> 4;

  __shared__ float Xs[16 * XS_STRIDE];
  __shared__ float Sls[16 * SL_STRIDE];

  // stage X tile [16 x 512]
  {
    const float* xsrc = x + ((size_t)b * SEQ_T + t0) * D_MODEL;
    for (int idx = tid; idx < 16 * 512; idx += 256) {
      int rr = idx >> 9, cc = idx & 511;
      Xs[rr * XS_STRIDE + cc] = xsrc[(size_t)rr * D_MODEL + cc];
    }
  }
  __syncthreads();

  const int n0 = wave * 64;  // this wave's 64-column slice of D
  v8f acc[4] = {};

  // ---- phase 1: acc = Xtile @ (W_base + W_current)^T ----
  for (int kb = 0; kb < D_MODEL; kb += 4) {
    v2f a = *(const v2f*)(&Xs[r * XS_STRIDE + kb + 2 * h]);
#pragma unroll
    for (int ct = 0; ct < 4; ++ct) {
      const size_t boff = (size_t)(n0 + ct * 16 + r) * D_MODEL + kb + 2 * h;
      v2f b1 = *(const v2f*)(Wb + boff);
      v2f b2 = *(const v2f*)(Wcur + boff);
      acc[ct] = wmma4(a, b1 + b2, acc[ct]);
    }
  }

  // ---- phase 2: causal attention  acc += mask(X@A^T) @ U ----
  const int sct = wave & 3;   // score col-tile this wave builds
  const int skh = wave >> 2;  // K-half (0: k<256, 1: k>=256)
  for (int sb = 0; sb <= t0; sb += 64) {
    // prefetch next s-block (A rows for scores, U rows for PV)
    {
      int pa = sb + 64 + sct * 16 + r;
      if (pa > SEQ_T - 1) pa = SEQ_T - 1;
      __builtin_prefetch(A + (size_t)pa * D_MODEL + skh * 256, 0, 1);
      int pu = sb + 64 + lane * 2;
      if (pu > SEQ_T - 1) pu = SEQ_T - 1;
      __builtin_prefetch(U + (size_t)pu * D_MODEL + n0, 0, 1);
    }
    // partial scores S[16 x 64] = Xtile @ A[sb:sb+64]^T (split over waves)
    v8f sacc = {};
    const float* arow = A + (size_t)(sb + sct * 16 + r) * D_MODEL;
    for (int kb = skh * 256; kb < skh * 256 + 256; kb += 4) {
      v2f a = *(const v2f*)(&Xs[r * XS_STRIDE + kb + 2 * h]);
      v2f bb = *(const v2f*)(arow + kb + 2 * h);
      sacc = wmma4(a, bb, sacc);
    }
    __syncthreads();  // previous Sls consumers done
    if (skh == 0) {
#pragma unroll
      for (int v = 0; v < 8; ++v)
        Sls[(v + 8 * h) * SL_STRIDE + sct * 16 + r] = sacc[v];
    }
    __syncthreads();
    if (skh == 1) {
#pragma unroll
      for (int v = 0; v < 8; ++v)
        Sls[(v + 8 * h) * SL_STRIDE + sct * 16 + r] += sacc[v];
    }
    __syncthreads();

    // PV: acc += S @ U[sb:sb+64][n0:n0+64], strict causal mask on last block
    const bool diag = (sb + 64 > t0);
    for (int kk = 0; kk < 64; kk += 4) {
      const int ks0 = kk + 2 * h;
      v2f a;
      a.x = Sls[r * SL_STRIDE + ks0];
      a.y = Sls[r * SL_STRIDE + ks0 + 1];
      if (diag) {
        if (sb + ks0 >= t0 + r) a.x = 0.f;  // s >= t -> excluded
        if (sb + ks0 + 1 >= t0 + r) a.y = 0.f;
      }
#pragma unroll
      for (int ct = 0; ct < 4; ++ct) {
        const float* ucol = U + (size_t)(sb + ks0) * D_MODEL + n0 + ct * 16 + r;
        v2f bb;
        bb.x = ucol[0];
        bb.y = ucol[D_MODEL];
        acc[ct] = wmma4(a, bb, acc[ct]);
      }
    }
  }

  // ---- phase 3: gate = sigmoid(Xtile @ gate_w^T + gate_b); out = gate*acc ----
  v8f gacc[4] = {};
  for (int kb = 0; kb < D_MODEL; kb += 4) {
    v2f a = *(const v2f*)(&Xs[r * XS_STRIDE + kb + 2 * h]);
#pragma unroll
    for (int ct = 0; ct < 4; ++ct) {
      const float* brow = gw + (size_t)(n0 + ct * 16 + r) * D_MODEL;
      v2f bb = *(const v2f*)(brow + kb + 2 * h);
      gacc[ct] = wmma4(a, bb, gacc[ct]);
    }
  }
#pragma unroll
  for (int ct = 0; ct < 4; ++ct) {
    const int col = n0 + ct * 16 + r;
    const float bias = gb[col];
#pragma unroll
    for (int v = 0; v < 8; ++v) {
      const int row = v + 8 * h;
      float g = 1.0f / (1.0f + expf(-(gacc[ct][v] + bias)));
      out[((size_t)b * SEQ_T + t0 + row) * D_MODEL + col] = g * acc[ct][v];
    }
  }
}

// ---------------------------------------------------------------------------
extern "C" void kernel_launch(void* const* d_in, const int* in_sizes, int n_in,
                              void* d_out, int out_size, void* d_ws,
                              size_t ws_size, hipStream_t stream) {
  (void)in_sizes; (void)n_in; (void)out_size; (void)ws_size;
  const float* x    = (const float*)d_in[0];
  const float* Wcur = (const float*)d_in[1];
  const float* Wb   = (const float*)d_in[2];
  const float* gw   = (const float*)d_in[3];
  const float* gb   = (const float*)d_in[4];
  float* out = (float*)d_out;

  float* ws = (float*)d_ws;
  float* A      = ws;                                  // T*D
  float* P0     = A + (size_t)SEQ_T * D_MODEL;         // T*D
  float* U      = P0 + (size_t)SEQ_T * D_MODEL;        // T*D
  float* anorm2 = U + (size_t)SEQ_T * D_MODEL;         // T
  float* Wc     = anorm2 + SEQ_T;                      // D*D
  float* Wfinal = out + (size_t)BATCH * SEQ_T * D_MODEL;

  k_avg<<<dim3(SEQ_T), dim3(512), 0, stream>>>(x, A, anorm2);
  k_p0<<<dim3(SEQ_T / 16, D_MODEL / 64), dim3(128), 0, stream>>>(A, Wb, P0);
  k_scan<<<dim3(1), dim3(1024), 0, stream>>>(Wcur, A, P0, anorm2, Wc, U, Wfinal);
  k_attn<<<dim3(SEQ_T / 16, BATCH), dim3(256), 0, stream>>>(x, Wb, Wcur, gw, gb,
                                                            A, U, out);
}